// SimpleGNN_28999619183030
// MI455X (gfx1250) — compile-verified
//
#include <hip/hip_runtime.h>

#define N_NODES     100000
#define N_EDGES     3200000
#define NUM_NUMERIC 32
#define N_CAT       6
#define VOCAB       1000
#define EMBED_DIM   16
#define IN_DIM      128   // 32 + 6*16
#define HIDDEN      64

typedef __attribute__((ext_vector_type(2))) float v2f;
typedef __attribute__((ext_vector_type(8))) float v8f;

// workspace layout (floats):
//   h   : N_NODES*HIDDEN
//   agg : N_NODES*HIDDEN
//   deg : N_NODES            (contiguous after agg -> zeroed together)

__global__ void zero_kernel(float* __restrict__ p, long n) {
    long i      = (long)blockIdx.x * blockDim.x + threadIdx.x;
    long stride = (long)gridDim.x * blockDim.x;
    for (; i < n; i += stride) p[i] = 0.0f;
}

// ---------------------------------------------------------------------------
// Fused feature-build + input GEMM + bias + ReLU, fp32 WMMA 16x16x4.
//   block = 128 threads = 4 waves; wave w owns output columns [16w, 16w+16).
//   Per M-tile (16 nodes): 32 K-steps of v_wmma_f32_16x16x4_f32 per wave.
// ---------------------------------------------------------------------------
#define XS 132   // X tile LDS row stride (pad: stride%64==4 -> conflict-free A reads)
#define WSD 80   // W LDS row stride (pad: half-wave bank sets disjoint)

__global__ __launch_bounds__(128)
void gemm_in_kernel(const float* __restrict__ x_num,
                    const int*   __restrict__ x_cat,
                    const float* __restrict__ emb,
                    const float* __restrict__ W_in,
                    const float* __restrict__ b_in,
                    float*       __restrict__ h) {
    __shared__ float Wlds[IN_DIM * WSD];   // 40 KB
    __shared__ float Xlds[16 * XS];        // 8.25 KB

    const int tid = threadIdx.x;

    // Stage W_in (128x64) into LDS once per block.
    for (int i = tid; i < IN_DIM * HIDDEN; i += 128) {
        int k = i >> 6, n = i & 63;
        Wlds[k * WSD + n] = W_in[i];
    }

    const int lane = tid & 31;
    const int wv   = tid >> 5;                 // 0..3 : N-tile
    const int ncol = wv * 16 + (lane & 15);    // output column 0..63
    const int koff = (lane >> 4) << 1;         // 0 (lanes 0-15) or 2 (lanes 16-31)
    const int arow = lane & 15;                // A-matrix row M
    const int hi   = lane >> 4;                // C layout: M = r + 8*hi
    const float bias = b_in[ncol];

    const int NTILES = N_NODES / 16;           // 6250 exactly
    for (int mt = blockIdx.x; mt < NTILES; mt += gridDim.x) {
        const int mbase = mt * 16;
        __syncthreads();   // previous tile's LDS reads done

        // Build X tile: this thread owns feature f = tid for all 16 nodes.
        {
            const int f = tid;  // 0..127
            if (f < NUM_NUMERIC) {
                for (int nl = 0; nl < 16; ++nl)
                    Xlds[nl * XS + f] = x_num[(mbase + nl) * NUM_NUMERIC + f];
            } else {
                const int c = (f - NUM_NUMERIC) >> 4;
                const int e = (f - NUM_NUMERIC) & 15;
                const float* embc = emb + ((long)c * VOCAB) * EMBED_DIM + e;
                for (int nl = 0; nl < 16; ++nl) {
                    int idx = x_cat[(mbase + nl) * N_CAT + c];
                    Xlds[nl * XS + f] = embc[(long)idx * EMBED_DIM];
                }
            }
        }
        __syncthreads();

        v8f acc;
        #pragma unroll
        for (int r = 0; r < 8; ++r) acc[r] = bias;   // C[r] has N = lane%16

        #pragma unroll
        for (int kk = 0; kk < IN_DIM; kk += 4) {
            v2f a, b;
            // A 16x4 fp32 layout: lanes0-15 -> K=kk,kk+1 ; lanes16-31 -> K=kk+2,kk+3
            a.x = Xlds[arow * XS + kk + koff];
            a.y = Xlds[arow * XS + kk + koff + 1];
            // B 4x16 fp32 layout (row-striped across lanes, same half-wave K split)
            b.x = Wlds[(kk + koff)     * WSD + ncol];
            b.y = Wlds[(kk + koff + 1) * WSD + ncol];
            acc = __builtin_amdgcn_wmma_f32_16x16x4_f32(
                      false, a, false, b, (short)0, acc, false, false);
        }

        // ReLU + store per C layout: VGPR r -> node mbase + r + 8*hi, col ncol.
        #pragma unroll
        for (int r = 0; r < 8; ++r) {
            float v = acc[r] > 0.0f ? acc[r] : 0.0f;
            h[(long)(mbase + r + hi * 8) * HIDDEN + ncol] = v;
        }
    }
}

// ---------------------------------------------------------------------------
// Edge scatter-add: one wave32 per edge; lane covers features f and f+32.
// h/agg are L2-resident (25.6 MB each << 192 MB L2) so random traffic stays
// on-chip; f32 atomics without return ride STOREcnt.
// ---------------------------------------------------------------------------
__global__ __launch_bounds__(256)
void edge_agg_kernel(const int*   __restrict__ ei,
                     const float* __restrict__ h,
                     float*       __restrict__ agg,
                     float*       __restrict__ deg) {
    const int lane   = threadIdx.x & 31;
    const int wave   = blockIdx.x * (blockDim.x >> 5) + (threadIdx.x >> 5);
    const int nwaves = gridDim.x * (blockDim.x >> 5);

    for (int e = wave; e < N_EDGES; e += nwaves) {
        const int s = ei[e];                 // src (wave-uniform -> scalar load)
        const int d = ei[N_EDGES + e];       // dst

        const int en = e + nwaves;           // prefetch next source row
        if (en < N_EDGES)
            __builtin_prefetch(h + (long)ei[en] * HIDDEN, 0, 0);

        const float* hs = h + (long)s * HIDDEN;
        const float v0 = hs[lane];
        const float v1 = hs[lane + 32];
        float* ad = agg + (long)d * HIDDEN;
        atomicAdd(ad + lane,      v0);
        atomicAdd(ad + lane + 32, v1);
        if (lane == 0) atomicAdd(deg + d, 1.0f);
    }
}

// ---------------------------------------------------------------------------
// Normalize by clipped degree, ReLU, dot with W_out, +b_out.
// One wave32 per node; 5-step shuffle reduction (wave32!).
// ---------------------------------------------------------------------------
__global__ __launch_bounds__(256)
void out_kernel(const float* __restrict__ agg,
                const float* __restrict__ deg,
                const float* __restrict__ W_out,
                const float* __restrict__ b_out,
                float*       __restrict__ out) {
    const int lane   = threadIdx.x & 31;
    const int wave   = blockIdx.x * (blockDim.x >> 5) + (threadIdx.x >> 5);
    const int nwaves = gridDim.x * (blockDim.x >> 5);

    const float w0 = W_out[lane];
    const float w1 = W_out[lane + 32];
    const float b  = b_out[0];

    for (int n = wave; n < N_NODES; n += nwaves) {
        float dg  = deg[n];
        dg        = dg < 1.0f ? 1.0f : dg;     // clip(deg, 1.0)
        const float inv = 1.0f / dg;
        float a0 = agg[(long)n * HIDDEN + lane]      * inv;
        float a1 = agg[(long)n * HIDDEN + lane + 32] * inv;
        a0 = a0 > 0.0f ? a0 : 0.0f;            // ReLU after normalize
        a1 = a1 > 0.0f ? a1 : 0.0f;
        float p = a0 * w0 + a1 * w1;
        #pragma unroll
        for (int off = 16; off > 0; off >>= 1)
            p += __shfl_down(p, off, 32);
        if (lane == 0) out[n] = p + b;
    }
}

extern "C" void kernel_launch(void* const* d_in, const int* in_sizes, int n_in,
                              void* d_out, int out_size, void* d_ws, size_t ws_size,
                              hipStream_t stream) {
    const float* x_num = (const float*)d_in[0];
    const int*   x_cat = (const int*)  d_in[1];
    const int*   ei    = (const int*)  d_in[2];
    const float* emb   = (const float*)d_in[3];
    const float* W_in  = (const float*)d_in[4];
    const float* b_in  = (const float*)d_in[5];
    const float* W_out = (const float*)d_in[6];
    const float* b_out = (const float*)d_in[7];
    float* out = (float*)d_out;

    float* h   = (float*)d_ws;
    float* agg = h   + (size_t)N_NODES * HIDDEN;
    float* deg = agg + (size_t)N_NODES * HIDDEN;
    (void)deg; (void)in_sizes; (void)n_in; (void)out_size; (void)ws_size;

    // agg and deg are contiguous: zero both in one pass.
    const long nz = (long)N_NODES * HIDDEN + N_NODES;
    zero_kernel   <<<2048, 256, 0, stream>>>(agg, nz);
    gemm_in_kernel<<<1024, 128, 0, stream>>>(x_num, x_cat, emb, W_in, b_in, h);
    edge_agg_kernel<<<4096, 256, 0, stream>>>(ei, h, agg, deg);
    out_kernel    <<<1024, 256, 0, stream>>>(agg, deg, W_out, b_out, out);
}